// RotatedIoU3DLoss_36696200577661
// MI455X (gfx1250) — compile-verified
//
#include <hip/hip_runtime.h>
#include <hip/hip_bf16.h>

typedef float v2f __attribute__((ext_vector_type(2)));
typedef float v8f __attribute__((ext_vector_type(8)));

#define TPB 256
#define NBLK 1024

// Strictly monotone surrogate for atan2(y,x): pseudo-angle in (-2, 2].
// Sort order is identical to atan2 (used only as an argsort key in the
// reference); ±0 edge cases map consistently (atan2(±0,-x)=±pi <-> ±2,
// atan2(±0,+x)=±0 <-> ±0, equal keys compare equal either way).
__device__ __forceinline__ float pseudo_angle(float y, float x) {
    float d = fabsf(x) + fabsf(y);
    if (d == 0.0f) return 0.0f;          // atan2(0,0) == 0
    return copysignf(1.0f - x / d, y);
}

__device__ __forceinline__ float pair_loss(const float* __restrict__ b1,
                                           const float* __restrict__ b2) {
    const float EPS = 1e-8f;
    // 2D box params: indices [0,1,3,4,6] of the 7-dof box
    float x1 = b1[0], y1 = b1[1], w1 = b1[3], h1 = b1[4], a1 = b1[6];
    float x2 = b2[0], y2 = b2[1], w2 = b2[3], h2 = b2[4], a2 = b2[6];
    float c1 = cosf(a1), s1 = sinf(a1);
    float c2 = cosf(a2), s2 = sinf(a2);

    const float xm[4] = {0.5f, -0.5f, -0.5f, 0.5f};
    const float ym[4] = {0.5f, 0.5f, -0.5f, -0.5f};
    float C1x[4], C1y[4], C2x[4], C2y[4];
#pragma unroll
    for (int k = 0; k < 4; ++k) {
        float xa = xm[k] * w1, ya = ym[k] * h1;
        C1x[k] = xa * c1 - ya * s1 + x1;
        C1y[k] = xa * s1 + ya * c1 + y1;
        float xb = xm[k] * w2, yb = ym[k] * h2;
        C2x[k] = xb * c2 - yb * s2 + x2;
        C2y[k] = xb * s2 + yb * c2 + y2;
    }

    // 24 candidate vertices: 0-3 corners1, 4-7 corners2, 8-23 edge intersections
    float Px[24], Py[24];
    bool  Mk[24];

    // corners1 inside box2
    {
        float ax = C2x[0], ay = C2y[0];
        float abx = C2x[1] - ax, aby = C2y[1] - ay;
        float adx = C2x[3] - ax, ady = C2y[3] - ay;
        float iab = 1.0f / (abx * abx + aby * aby);
        float iad = 1.0f / (adx * adx + ady * ady);
#pragma unroll
        for (int k = 0; k < 4; ++k) {
            float amx = C1x[k] - ax, amy = C1y[k] - ay;
            float pab = (amx * abx + amy * aby) * iab;
            float pad = (amx * adx + amy * ady) * iad;
            const float e = 1e-6f;
            Mk[k] = (pab > -e) & (pab < 1.0f + e) & (pad > -e) & (pad < 1.0f + e);
            Px[k] = C1x[k];
            Py[k] = C1y[k];
        }
    }
    // corners2 inside box1
    {
        float ax = C1x[0], ay = C1y[0];
        float abx = C1x[1] - ax, aby = C1y[1] - ay;
        float adx = C1x[3] - ax, ady = C1y[3] - ay;
        float iab = 1.0f / (abx * abx + aby * aby);
        float iad = 1.0f / (adx * adx + ady * ady);
#pragma unroll
        for (int k = 0; k < 4; ++k) {
            float amx = C2x[k] - ax, amy = C2y[k] - ay;
            float pab = (amx * abx + amy * aby) * iab;
            float pad = (amx * adx + amy * ady) * iad;
            const float e = 1e-6f;
            Mk[4 + k] = (pab > -e) & (pab < 1.0f + e) & (pad > -e) & (pad < 1.0f + e);
            Px[4 + k] = C2x[k];
            Py[4 + k] = C2y[k];
        }
    }
    // segment-segment intersections (strict (0,1) on both params, EPS on denom)
#pragma unroll
    for (int i = 0; i < 4; ++i) {
        int in = (i + 1) & 3;
        float p1x = C1x[i], p1y = C1y[i];
        float rx = C1x[in] - p1x, ry = C1y[in] - p1y;
#pragma unroll
        for (int j = 0; j < 4; ++j) {
            int jn = (j + 1) & 3;
            float p2x = C2x[j], p2y = C2y[j];
            float sx = C2x[jn] - p2x, sy = C2y[jn] - p2y;
            float den = rx * sy - ry * sx;
            float dpx = p2x - p1x, dpy = p2y - p1y;
            float invden = 1.0f / (den + EPS);
            float t = (dpx * sy - dpy * sx) * invden;
            float u = (dpx * ry - dpy * rx) * invden;
            bool m = (t > 0.0f) & (t < 1.0f) & (u > 0.0f) & (u < 1.0f);
            int idx = 8 + i * 4 + j;
            Mk[idx] = m;
            Px[idx] = m ? (p1x + t * rx) : 0.0f;
            Py[idx] = m ? (p1y + t * ry) : 0.0f;
        }
    }

    // centroid of valid vertices (divide by max(nv,1) like reference)
    float sx_ = 0.0f, sy_ = 0.0f;
    int nv = 0;
#pragma unroll
    for (int k = 0; k < 24; ++k) {
        if (Mk[k]) { sx_ += Px[k]; sy_ += Py[k]; ++nv; }
    }
    float dn = (float)(nv > 1 ? nv : 1);
    float mx = sx_ / dn, my = sy_ / dn;

    // compact valid vertices and insertion-sort by pseudo-angle
    // (stable -> matches jnp.argsort stable order on the same concatenation)
    float Ang[24], Vx[24], Vy[24];
    int cnt = 0;
    for (int k = 0; k < 24; ++k) {
        if (Mk[k]) {
            float vx = Px[k] - mx, vy = Py[k] - my;
            float an = pseudo_angle(vy, vx);
            int p = cnt - 1;
            while (p >= 0 && Ang[p] > an) {
                Ang[p + 1] = Ang[p]; Vx[p + 1] = Vx[p]; Vy[p + 1] = Vy[p];
                --p;
            }
            Ang[p + 1] = an; Vx[p + 1] = vx; Vy[p + 1] = vy;
            ++cnt;
        }
    }

    // shoelace over sorted ring (invalid entries are zero in reference and drop out)
    float total = 0.0f;
    for (int k = 0; k + 1 < cnt; ++k)
        total += Vx[k] * Vy[k + 1] - Vx[k + 1] * Vy[k];
    if (cnt > 0)
        total += Vx[cnt - 1] * Vy[0] - Vx[0] * Vy[cnt - 1];
    float area2d = fabsf(total) * 0.5f;

    // z extent overlap + IoU
    float z1 = b1[2], l1 = b1[5], z2 = b2[2], l2 = b2[5];
    float zo = fminf(z1 + 0.5f * l1, z2 + 0.5f * l2) -
               fmaxf(z1 - 0.5f * l1, z2 - 0.5f * l2);
    zo = fmaxf(zo, 0.0f);
    float inter3d = area2d * zo;
    float vol1 = b1[3] * b1[4] * b1[5];
    float vol2 = b2[3] * b2[4] * b2[5];
    float iou = inter3d / (vol1 + vol2 - inter3d);
    return 1.0f - iou;
}

__global__ void __launch_bounds__(TPB)
rotated_iou3d_loss_kernel(const float* __restrict__ pred,
                          const float* __restrict__ tgt,
                          float* __restrict__ partial, int N) {
    __shared__ float sp[TPB * 7];
    __shared__ float st[TPB * 7];
    __shared__ float wsum[TPB / 32];

    float acc = 0.0f;
    const long long step = (long long)gridDim.x * TPB;

    for (long long base = (long long)blockIdx.x * TPB; base < N; base += step) {
        long long off = base * 7;
        long long rem = (long long)N - base;
        int chunk = (int)((rem < TPB ? rem : TPB) * 7);

        __syncthreads();  // previous iteration done reading LDS
        for (int k = threadIdx.x; k < chunk; k += TPB) {
            sp[k] = pred[off + k];
            st[k] = tgt[off + k];
        }
        // prefetch next chunk while this one is processed (global_prefetch_b8)
        long long nbase = base + step;
        if (nbase + (long long)threadIdx.x < N) {
            __builtin_prefetch(&pred[nbase * 7 + (long long)threadIdx.x * 7], 0, 0);
            __builtin_prefetch(&tgt[nbase * 7 + (long long)threadIdx.x * 7], 0, 0);
        }
        __syncthreads();

        long long i = base + threadIdx.x;
        if (i < N)
            acc += pair_loss(&sp[threadIdx.x * 7], &st[threadIdx.x * 7]);
    }

    // --- wave-level reduction via full-fp32 WMMA (exact: x*1.0 products) ---
    // A (16x4 f32, 2 VGPRs/lane) holds the wave's 32 partials (second slot 0);
    // B = all-ones; D rows become row-sums replicated across columns.
    v2f a; a.x = acc; a.y = 0.0f;
    v2f b; b.x = 1.0f; b.y = 1.0f;
    v8f c = {};
    c = __builtin_amdgcn_wmma_f32_16x16x4_f32(
        /*neg_a=*/false, a, /*neg_b=*/false, b,
        /*c_mod=*/(short)0, c, /*reuse_a=*/false, /*reuse_b=*/false);
    float s = c[0] + c[1] + c[2] + c[3] + c[4] + c[5] + c[6] + c[7];
    // lanes 0-15 hold sum of rows 0-7, lanes 16-31 hold rows 8-15 -> combine
    s += __shfl_xor(s, 16, 32);

    int lane = threadIdx.x & 31;
    int wid = threadIdx.x >> 5;
    if (lane == 0) wsum[wid] = s;
    __syncthreads();
    if (threadIdx.x == 0) {
        float t = 0.0f;
#pragma unroll
        for (int w = 0; w < TPB / 32; ++w) t += wsum[w];
        partial[blockIdx.x] = t;
    }
}

__global__ void __launch_bounds__(256)
finalize_kernel(const float* __restrict__ partial, int nblk, int N,
                float* __restrict__ out) {
    __shared__ float sm[256];
    float t = 0.0f;
    for (int i = threadIdx.x; i < nblk; i += 256) t += partial[i];
    sm[threadIdx.x] = t;
    __syncthreads();
    for (int s = 128; s > 0; s >>= 1) {
        if (threadIdx.x < s) sm[threadIdx.x] += sm[threadIdx.x + s];
        __syncthreads();
    }
    if (threadIdx.x == 0) out[0] = sm[0] / (float)N;
}

extern "C" void kernel_launch(void* const* d_in, const int* in_sizes, int n_in,
                              void* d_out, int out_size, void* d_ws, size_t ws_size,
                              hipStream_t stream) {
    const float* pred = (const float*)d_in[0];
    const float* tgt  = (const float*)d_in[1];
    float* out = (float*)d_out;
    float* partial = (float*)d_ws;  // NBLK floats of scratch

    int N = in_sizes[0] / 7;

    rotated_iou3d_loss_kernel<<<NBLK, TPB, 0, stream>>>(pred, tgt, partial, N);
    finalize_kernel<<<1, 256, 0, stream>>>(partial, NBLK, N, out);
}